// WARP_CTC_LOSS_16312285790338
// MI455X (gfx1250) — compile-verified
//
#include <hip/hip_runtime.h>
#include <stdint.h>

// ---------------------------------------------------------------------------
// CTC log-likelihood (warp-ctc math), MI455X / gfx1250.
//   B=64, T=1024, V=512, Lmax=128, S=257, blank=0.
// Phase A: per-(b,t) row softmax normalizer + gather of the 129 reachable
//          classes into compact rows (bandwidth phase, 65536 blocks,
//          non-temporal logits streaming; compact rows stay L2-resident).
// Phase B: per-utterance serial alpha DP, one wave32 per b, states in
//          registers, neighbor exchange via shuffles. Compact rows are
//          streamed into LDS by the Tensor Data Mover: one tensor_load_to_lds
//          per 8 timesteps into a 4-deep LDS ring, synchronized with
//          s_wait_tensorcnt. Zero barriers on the critical path.
// Phase C: deterministic 64->1 reduction, mean, single float out.
// ---------------------------------------------------------------------------

#define NEGV (-1e30f)

typedef uint32_t u32x4 __attribute__((ext_vector_type(4)));
typedef uint32_t u32x8 __attribute__((ext_vector_type(8)));

__device__ __forceinline__ float logaddf(float x, float y) {
  float mx = fmaxf(x, y);
  float mn = fminf(x, y);
  // exp underflows to 0 for unreachable states; v_log_f32(1.0) == 0 exactly.
  return mx + __logf(1.0f + __expf(mn - mx));
}

// ---- CDNA5 Tensor Data Mover: load nrows x 132 f32 rows into LDS ----------
// D# per CDNA5 ISA. 2D tensor (VADDR2/3 = NULL).
//   group0: count=1 | lds_addr | global_addr[56:0] | type=2
//   group1: data_size=4B, tensor_dim0=132, tensor_dim1=nrows,
//           tile_dim0=132, tile_dim1=nrows, tensor_dim0_stride=132
// tile == tensor extent per call -> no OOB path. Tracked on TENSORcnt.
__device__ __forceinline__ void tdm_load_rows(const float* gsrc, void* lds_dst,
                                              int nrows) {
  uint64_t ga = (uint64_t)(uintptr_t)gsrc;
  uint32_t la = (uint32_t)(uintptr_t)lds_dst;   // low 32 bits = LDS byte offset
  u32x4 g0;
  g0[0] = 1u;                                   // count=1, user mode
  g0[1] = la;                                   // lds_addr
  g0[2] = (uint32_t)ga;                         // global_addr[31:0]
  g0[3] = (uint32_t)(ga >> 32) | (2u << 30);    // global_addr[56:32], type=2
  u32x8 g1;
  g1[0] = 2u << 16;                             // data_size=2 (4 bytes)
  g1[1] = 132u << 16;                           // tensor_dim0[15:0]=132
  g1[2] = (uint32_t)nrows << 16;                // dim0 hi=0 | tensor_dim1=nrows
  g1[3] = 132u << 16;                           // dim1 hi=0 | tile_dim0=132
  g1[4] = (uint32_t)nrows;                      // tile_dim1=nrows, tile_dim2=0
  g1[5] = 132u;                                 // tensor_dim0_stride[31:0]=132
  g1[6] = 0u;                                   // stride0 hi | stride1 lo
  g1[7] = 0u;                                   // stride1 hi
  asm volatile("tensor_load_to_lds %0, %1" :: "s"(g0), "s"(g1) : "memory");
}

// ---------------------------------------------------------------------------
// Phase A: one block per (b,t) row. 256 threads, 2 elems/thread.
// Compact row layout (stride 132 floats): c[0]=blank logp, c[1..128]=labels.
// ---------------------------------------------------------------------------
__global__ __launch_bounds__(256) void ctc_rownorm_kernel(
    const float* __restrict__ logits,   // [B*T*512]
    const int*   __restrict__ labels,   // [B*128]
    float*       __restrict__ cOut)     // [B*T*132]
{
  const int row = blockIdx.x;           // b*1024 + t
  const int b   = row >> 10;
  const int tid = threadIdx.x;

  __shared__ float xs[512];
  __shared__ float mPart[8], sPart[8];

  const float* xrow = logits + (size_t)row * 512;
  // Stream logits non-temporally: 128 MB read exactly once; keep L2 for
  // the compact rows Phase B re-reads.
  float x0 = __builtin_nontemporal_load(&xrow[tid]);
  float x1 = __builtin_nontemporal_load(&xrow[tid + 256]);
  xs[tid] = x0; xs[tid + 256] = x1;

  // Online (max, sum-exp) pair, combined in one reduction pass.
  float m = fmaxf(x0, x1);
  float s = __expf(x0 - m) + __expf(x1 - m);
#pragma unroll
  for (int off = 16; off >= 1; off >>= 1) {
    float mo = __shfl_xor(m, off, 32);
    float so = __shfl_xor(s, off, 32);
    float M  = fmaxf(m, mo);
    s = s * __expf(m - M) + so * __expf(mo - M);
    m = M;
  }
  const int wave = tid >> 5;
  if ((tid & 31) == 0) { mPart[wave] = m; sPart[wave] = s; }
  __syncthreads();

  float M = mPart[0], S = sPart[0];
#pragma unroll
  for (int w = 1; w < 8; ++w) {
    float mo = mPart[w], so = sPart[w];
    float Mn = fmaxf(M, mo);
    S = S * __expf(M - Mn) + so * __expf(mo - Mn);
    M = Mn;
  }
  const float norm = M + __logf(S);   // log-sum-exp of the row

  if (tid < 129) {
    int idx = (tid == 0) ? 0 : labels[b * 128 + (tid - 1)];
    cOut[(size_t)row * 132 + tid] = xs[idx] - norm;
  }
}

// ---------------------------------------------------------------------------
// Phase B: one wave32 per utterance. Lane L owns states s = k*32+L, k=0..7;
// state 256 is a scalar slot. Compact rows arrive in 8-row chunks via the
// TDM into a 4-deep LDS ring (prefetch depth = 3 chunks = 24 timesteps).
// ---------------------------------------------------------------------------
__global__ __launch_bounds__(32) void ctc_dp_kernel(
    const float* __restrict__ cIn,      // [B*1024*132]
    const int*   __restrict__ labels,   // [B*128]
    const int*   __restrict__ in_len,   // [B]
    const int*   __restrict__ lab_len,  // [B]
    float*       __restrict__ lls)      // [B]
{
  const int b = blockIdx.x;
  const int L = threadIdx.x;            // lane id, wave32

  __shared__ float cring[4][8][132];    // TDM ring: 4 chunks x 8 rows
  __shared__ float aout[257];

  const int Tb = in_len[b];
  const int Lb = lab_len[b];
  const float* crow = cIn + (size_t)b * 1024 * 132;
  const int*   lab  = labels + b * 128;

  // Per-slot constants: compact index and skip-transition legality.
  int  idx[8];
  bool skip[8];
#pragma unroll
  for (int k = 0; k < 8; ++k) {
    int  s   = k * 32 + L;
    bool odd = (s & 1) != 0;
    idx[k] = odd ? ((s + 1) >> 1) : 0;                       // labels vs blank
    bool sk = false;
    if (odd && s >= 3) sk = (lab[(s - 1) >> 1] != lab[(s - 3) >> 1]);
    skip[k] = sk;
  }

  const int nc = (Tb + 7) >> 3;         // number of 8-row chunks
  int ci = 0;                           // chunks issued so far
  for (; ci < nc && ci < 3; ++ci) {     // prologue: prefetch 3 chunks
    const int t0 = ci * 8;
    const int nr = min(8, Tb - t0);
    tdm_load_rows(crow + (size_t)t0 * 132, &cring[ci & 3][0][0], nr);
  }

  float a[8], a8 = NEGV;
#pragma unroll
  for (int k = 0; k < 8; ++k) a[k] = NEGV;

  for (int c = 0; c < nc; ++c) {
    // TDM completes in order per wave: wait until chunk c has landed.
    const int pending = ci - c - 1;     // newer chunks allowed in flight
    if (pending >= 2)      asm volatile("s_wait_tensorcnt 0x2" ::: "memory");
    else if (pending == 1) asm volatile("s_wait_tensorcnt 0x1" ::: "memory");
    else                   asm volatile("s_wait_tensorcnt 0x0" ::: "memory");

    const int slot = c & 3;
    const int t0   = c * 8;
    const int tend = min(t0 + 8, Tb);
    for (int t = t0; t < tend; ++t) {
      const float* cr = &cring[slot][t - t0][0];
      if (t == 0) {
        // alpha0: only states 0 (blank) and 1 (first label) live.
        float c0 = cr[0], c1 = cr[1];
        if (L == 0) a[0] = c0;
        if (L == 1) a[0] = c1;
        continue;
      }
      const float cblank = cr[0];
      float an[8];
#pragma unroll
      for (int k = 0; k < 8; ++k) {
        // alpha[s-1]
        float t1  = __shfl_up(a[k], 1, 32);
        float c1v = __shfl(a[k == 0 ? 7 : k - 1], 31, 32);
        float a1  = (L == 0) ? ((k == 0) ? NEGV : c1v) : t1;
        // alpha[s-2] (only if skip legal)
        float t2  = __shfl_up(a[k], 2, 32);
        float c2v = __shfl(a[k == 0 ? 7 : k - 1], 30 + L, 32);
        float a2  = skip[k] ? ((L < 2) ? ((k == 0) ? NEGV : c2v) : t2) : NEGV;

        float lp = cr[idx[k]];
        an[k] = logaddf(logaddf(a[k], a1), a2) + lp;
      }
      // state 256 (even, blank, no skip); uses OLD a[7].
      float prev255 = __shfl(a[7], 31, 32);
      a8 = logaddf(a8, prev255) + cblank;
#pragma unroll
      for (int k = 0; k < 8; ++k) a[k] = an[k];
    }

    // Refill the ring slot just freed (chunk c+3 overwrites slot (c+3)&3,
    // which held chunk c-1: fully consumed).
    if (ci < nc) {
      const int t0n = ci * 8;
      const int nr  = min(8, Tb - t0n);
      tdm_load_rows(crow + (size_t)t0n * 132, &cring[ci & 3][0][0], nr);
      ++ci;
    }
  }

  // Readout: logaddexp(alpha[2L-1], alpha[2L]) via LDS (single wave, in-order).
#pragma unroll
  for (int k = 0; k < 8; ++k) aout[k * 32 + L] = a[k];
  if (L == 0) aout[256] = a8;
  __syncthreads();  // single-wave workgroup: S_NOP, kept for clarity
  if (L == 0) {
    lls[b] = logaddf(aout[2 * Lb - 1], aout[2 * Lb]);
  }
}

// ---------------------------------------------------------------------------
// Phase C: mean over B=64 log-likelihoods (deterministic shuffle tree).
// ---------------------------------------------------------------------------
__global__ __launch_bounds__(32) void ctc_finish_kernel(
    const float* __restrict__ lls, float* __restrict__ out)
{
  const int L = threadIdx.x;
  float v = lls[L] + lls[L + 32];
#pragma unroll
  for (int off = 16; off >= 1; off >>= 1) v += __shfl_xor(v, off, 32);
  if (L == 0) out[0] = v * (1.0f / 64.0f);
}

// ---------------------------------------------------------------------------
extern "C" void kernel_launch(void* const* d_in, const int* in_sizes, int n_in,
                              void* d_out, int out_size, void* d_ws, size_t ws_size,
                              hipStream_t stream) {
  const float* logits  = (const float*)d_in[0];   // [64,1024,512] f32
  const int*   labels  = (const int*)  d_in[1];   // [64,128] i32
  const int*   in_len  = (const int*)  d_in[2];   // [64] i32
  const int*   lab_len = (const int*)  d_in[3];   // [64] i32
  float*       out     = (float*)d_out;           // [1] f32

  // Workspace: [0,256)B -> lls[64]; [1024,...) -> compact rows (~34.6 MB).
  float* lls  = (float*)d_ws;
  float* cbuf = (float*)((char*)d_ws + 1024);

  (void)in_sizes; (void)n_in; (void)out_size; (void)ws_size;

  ctc_rownorm_kernel<<<64 * 1024, 256, 0, stream>>>(logits, labels, cbuf);
  ctc_dp_kernel     <<<64,        32,  0, stream>>>(cbuf, labels, in_len, lab_len, lls);
  ctc_finish_kernel <<<1,         32,  0, stream>>>(lls, out);
}